// GraphAttentionLayer_36867999269228
// MI455X (gfx1250) — compile-verified
//
#include <hip/hip_runtime.h>
#include <math.h>

#define N_NODES 100000
#define N_EDGES 1600000
#define D_IN    256
#define D_OUT   64
#define ALPHA   0.2f

typedef float v2f __attribute__((ext_vector_type(2)));
typedef float v8f __attribute__((ext_vector_type(8)));

// ---- order-preserving float <-> uint encoding for atomic max on floats ----
__device__ __forceinline__ unsigned fenc(float f) {
  unsigned b = __float_as_uint(f);
  return b ^ ((((int)b) >> 31) | 0x80000000u);
}
__device__ __forceinline__ float fdec(unsigned u) {
  unsigned mask = (((int)u) >= 0) ? 0xFFFFFFFFu : 0x80000000u;
  return __uint_as_float(u ^ mask);
}

// ============================================================================
// Kernel 0: init outputs / reduction buffers (harness poisons them to 0xAA)
// ============================================================================
__global__ __launch_bounds__(256)
void gat_init(float* __restrict__ out, unsigned* __restrict__ mkey,
              float* __restrict__ ssum) {
  int i = blockIdx.x * 256 + threadIdx.x;
  if (i < N_NODES * D_OUT) out[i] = 0.0f;
  if (i < N_NODES) { mkey[i] = 0u; ssum[i] = 0.0f; }  // 0 < fenc(any finite)
}

// ============================================================================
// Kernel 1: hw = h @ W via V_WMMA_F32_16X16X4_F32
//   block = 256 threads = 8 waves; each wave computes a 16x64 slab of hw.
//   W staged in LDS *pair-interleaved* so every WMMA B operand is exactly one
//   naturally-aligned ds_load_b64 into an even VGPR pair (no repack movs):
//     Wl[(k>>1)*128 + n*2 + (k&1)] = W[k][n]
// ============================================================================
__global__ __launch_bounds__(256)
void gat_gemm(const float* __restrict__ h, const float* __restrict__ W,
              float* __restrict__ hw) {
  __shared__ float Wl[D_IN * D_OUT];  // 64 KB (WGP has 320 KB)
  const int tid = threadIdx.x;

  // cooperative fill: float4 global loads, pair-interleaved LDS scatter
  const float4* Wv = (const float4*)W;
  #pragma unroll 4
  for (int idx = tid; idx < (D_IN * D_OUT) / 4; idx += 256) {
    const int k  = idx >> 4;          // K row 0..255
    const int n4 = (idx & 15) * 4;    // starting column of this float4
    const float4 w = Wv[idx];
    float* base = Wl + (k >> 1) * 128 + (k & 1);
    base[(n4 + 0) * 2] = w.x;
    base[(n4 + 1) * 2] = w.y;
    base[(n4 + 2) * 2] = w.z;
    base[(n4 + 3) * 2] = w.w;
  }
  __syncthreads();

  const int wave = tid >> 5;
  const int lane = tid & 31;
  const int m    = lane & 15;   // row within A tile / col within B,C,D tiles
  const int kg   = lane >> 4;   // K-group (0: K=0,1 ; 1: K=2,3) per ISA layout
  const int tile = blockIdx.x * 8 + wave;
  if (tile * 16 >= N_NODES) return;   // wave-uniform guard: EXEC stays all-1s
  const int r0 = tile * 16;

  v8f acc0 = {}, acc1 = {}, acc2 = {}, acc3 = {};

  const float* hrow  = h + (size_t)(r0 + m) * D_IN + 2 * kg;
  const float* bbase = Wl + kg * 128 + 2 * m;   // + (kb>>1)*128 per K-step

  #pragma unroll 4
  for (int kb = 0; kb < D_IN; kb += 4) {
    // A tile 16x4: lane holds A[m][2*kg + {0,1}]  -> one b64 global load
    v2f a = *(const v2f*)(hrow + kb);

    // B tile 4x16: pair {W[krow][n], W[krow+1][n]} -> one ds_load_b64 each
    const float* bp = bbase + (kb >> 1) * 128;
    v2f b0 = *(const v2f*)(bp +  0);   // cols  0..15
    v2f b1 = *(const v2f*)(bp + 32);   // cols 16..31
    v2f b2 = *(const v2f*)(bp + 64);   // cols 32..47
    v2f b3 = *(const v2f*)(bp + 96);   // cols 48..63

    acc0 = __builtin_amdgcn_wmma_f32_16x16x4_f32(false, a, false, b0, (short)0, acc0, false, false);
    acc1 = __builtin_amdgcn_wmma_f32_16x16x4_f32(false, a, false, b1, (short)0, acc1, false, false);
    acc2 = __builtin_amdgcn_wmma_f32_16x16x4_f32(false, a, false, b2, (short)0, acc2, false, false);
    acc3 = __builtin_amdgcn_wmma_f32_16x16x4_f32(false, a, false, b3, (short)0, acc3, false, false);
  }

  // C/D layout: VGPR v, lane -> row = 8*kg + v, col = 16*c + m
  float* orow = hw + (size_t)(r0 + kg * 8) * D_OUT + m;
  #pragma unroll
  for (int v = 0; v < 8; ++v) {
    orow[(size_t)v * D_OUT +  0] = acc0[v];
    orow[(size_t)v * D_OUT + 16] = acc1[v];
    orow[(size_t)v * D_OUT + 32] = acc2[v];
    orow[(size_t)v * D_OUT + 48] = acc3[v];
  }
}

// ============================================================================
// Kernel 2: per-node attention scalars  s1 = hw . a[0:64], s2 = hw . a[64:128]
//   one wave (32 lanes) per node; lane covers cols {lane, lane+32}
// ============================================================================
__global__ __launch_bounds__(256)
void gat_s12(const float* __restrict__ hw, const float* __restrict__ a,
             float* __restrict__ s1, float* __restrict__ s2) {
  const int wave = threadIdx.x >> 5;
  const int lane = threadIdx.x & 31;
  const int n = blockIdx.x * 8 + wave;
  if (n >= N_NODES) return;
  const float h0 = hw[(size_t)n * D_OUT + lane];
  const float h1 = hw[(size_t)n * D_OUT + 32 + lane];
  float p1 = h0 * a[lane]      + h1 * a[lane + 32];
  float p2 = h0 * a[64 + lane] + h1 * a[96 + lane];
  #pragma unroll
  for (int off = 16; off > 0; off >>= 1) {
    p1 += __shfl_xor(p1, off, 32);
    p2 += __shfl_xor(p2, off, 32);
  }
  if (lane == 0) { s1[n] = p1; s2[n] = p2; }
}

// ============================================================================
// Kernel 3: segment max over edges (numerically stable softmax prep)
// ============================================================================
__global__ __launch_bounds__(256)
void gat_edge_max(const float* __restrict__ s1, const float* __restrict__ s2,
                  const int* __restrict__ dst, const int* __restrict__ src,
                  unsigned* __restrict__ mkey) {
  int k = blockIdx.x * 256 + threadIdx.x;
  if (k >= N_EDGES) return;
  const int d = dst[k];
  float e = s1[d] + s2[src[k]];
  e = (e > 0.0f) ? e : ALPHA * e;                 // leaky relu
  atomicMax(&mkey[d], fenc(e));
}

// ============================================================================
// Kernel 4: ex = exp(e - m[dst]); segment sum into ssum[dst]
// ============================================================================
__global__ __launch_bounds__(256)
void gat_edge_exp(const float* __restrict__ s1, const float* __restrict__ s2,
                  const int* __restrict__ dst, const int* __restrict__ src,
                  const unsigned* __restrict__ mkey,
                  float* __restrict__ ex, float* __restrict__ ssum) {
  int k = blockIdx.x * 256 + threadIdx.x;
  if (k >= N_EDGES) return;
  const int d = dst[k];
  float e = s1[d] + s2[src[k]];
  e = (e > 0.0f) ? e : ALPHA * e;
  const float x = __expf(e - fdec(mkey[d]));
  ex[k] = x;
  atomicAdd(&ssum[d], x);
}

// ============================================================================
// Kernel 5: h' += att * hw[src]; one wave per edge, lanes cover 64 cols
//   (hw fits in the 192 MB L2, so gathers + f32 atomics stay on-chip)
// ============================================================================
__global__ __launch_bounds__(256)
void gat_aggregate(const float* __restrict__ hw, const int* __restrict__ dst,
                   const int* __restrict__ src, const float* __restrict__ ex,
                   const float* __restrict__ ssum, float* __restrict__ out) {
  const int wave = threadIdx.x >> 5;
  const int lane = threadIdx.x & 31;
  const int k = blockIdx.x * 8 + wave;
  if (k >= N_EDGES) return;
  const int d = dst[k];
  const int s = src[k];
  const float att = ex[k] / ssum[d];
  const float* hs = hw  + (size_t)s * D_OUT;
  float*       od = out + (size_t)d * D_OUT;
  atomicAdd(&od[lane],      att * hs[lane]);
  atomicAdd(&od[lane + 32], att * hs[lane + 32]);
}

// ============================================================================
// Kernel 6: final ELU in place
// ============================================================================
__global__ __launch_bounds__(256)
void gat_elu(float* __restrict__ out) {
  int i = blockIdx.x * 256 + threadIdx.x;
  if (i >= N_NODES * D_OUT) return;
  const float v = out[i];
  out[i] = (v > 0.0f) ? v : (__expf(v) - 1.0f);
}

// ============================================================================
extern "C" void kernel_launch(void* const* d_in, const int* in_sizes, int n_in,
                              void* d_out, int out_size, void* d_ws, size_t ws_size,
                              hipStream_t stream) {
  const float* h   = (const float*)d_in[0];   // [N, 256]
  const float* W   = (const float*)d_in[1];   // [256, 64]
  const float* a   = (const float*)d_in[2];   // [128, 1]
  const int*   dst = (const int*)  d_in[3];   // [E]
  const int*   src = (const int*)  d_in[4];   // [E]
  float* out = (float*)d_out;                 // [N, 64]

  // workspace layout (floats): hw | s1 | s2 | mkey | ssum | ex  (~33.6 MB)
  float*    hw   = (float*)d_ws;
  float*    s1   = hw  + (size_t)N_NODES * D_OUT;
  float*    s2   = s1  + N_NODES;
  unsigned* mkey = (unsigned*)(s2 + N_NODES);
  float*    ssum = (float*)(mkey + N_NODES);
  float*    ex   = ssum + N_NODES;

  const int ELEMS = N_NODES * D_OUT;

  gat_init     <<<(ELEMS + 255) / 256, 256, 0, stream>>>(out, mkey, ssum);
  gat_gemm     <<<(N_NODES / 16 + 7) / 8, 256, 0, stream>>>(h, W, hw);
  gat_s12      <<<(N_NODES + 7) / 8, 256, 0, stream>>>(hw, a, s1, s2);
  gat_edge_max <<<(N_EDGES + 255) / 256, 256, 0, stream>>>(s1, s2, dst, src, mkey);
  gat_edge_exp <<<(N_EDGES + 255) / 256, 256, 0, stream>>>(s1, s2, dst, src, mkey, ex, ssum);
  gat_aggregate<<<(N_EDGES + 7) / 8, 256, 0, stream>>>(hw, dst, src, ex, ssum, out);
  gat_elu      <<<(ELEMS + 255) / 256, 256, 0, stream>>>(out);
}